// LocalStyleAdaptor_72224170049933
// MI455X (gfx1250) — compile-verified
//
#include <hip/hip_runtime.h>
#include <hip/hip_bf16.h>

typedef __attribute__((ext_vector_type(2))) float v2f;
typedef __attribute__((ext_vector_type(8))) float v8f;

#define NB 16        // batches
#define NT 8192      // time steps
#define NH 256       // hidden
#define NP 512       // MAX_PH segments (output bins)
#define NM 64        // codebook size
#define NROWS (NB*NP)      // 8192 rows of x_flat
#define NTILES (NROWS/16)  // 512 WMMA row-tiles

// ---------------------------------------------------------------- init: zero histogram
__global__ void k_init(int* __restrict__ hist) {
    int i = threadIdx.x;
    if (i < NM) hist[i] = 0;
}

// ---------------------------------------------------------------- counts per (b, ph-1)
__global__ void k_cnt(const int* __restrict__ mel2ph, int* __restrict__ cnt) {
    __shared__ int h[NP];
    int b = blockIdx.x, tid = threadIdx.x;
    for (int i = tid; i < NP; i += 256) h[i] = 0;
    __syncthreads();
    for (int t = tid; t < NT; t += 256) {
        int ph = mel2ph[b * NT + t];
        if (ph > 0) atomicAdd(&h[ph - 1], 1);
    }
    __syncthreads();
    for (int i = tid; i < NP; i += 256) cnt[b * NP + i] = h[i];
}

// ---------------------------------------------------------------- segment sum (bulk 128 MiB pass)
// grid = 16 batches * 8 H-chunks of 32 cols; LDS accum 512*32 f32 = 64 KB
__global__ __launch_bounds__(256) void k_seg(const float* __restrict__ mels,
                                             const int* __restrict__ mel2ph,
                                             float* __restrict__ xsum) {
    __shared__ float acc[NP * 32];
    int b = blockIdx.x >> 3;
    int hc = blockIdx.x & 7;
    int tid = threadIdx.x;
    for (int i = tid; i < NP * 32; i += 256) acc[i] = 0.0f;
    __syncthreads();
    int w = tid >> 5, lane = tid & 31;
    const float* mb = mels + (size_t)b * NT * NH + hc * 32 + lane;
    const int* pb = mel2ph + b * NT;
    for (int t = w; t < NT; t += 8) {       // t is wave-uniform -> scalar ph load
        int ph = pb[t];
        if (ph > 0) {
            float v = mb[(size_t)t * NH];
            atomicAdd(&acc[(ph - 1) * 32 + lane], v);   // ds_add_f32
        }
    }
    __syncthreads();
    for (int i = tid; i < NP * 32; i += 256) {
        int p = i >> 5, c = i & 31;
        xsum[(size_t)(b * NP + p) * NH + hc * 32 + c] = acc[i];
    }
}

// ---------------------------------------------------------------- embedding squared norms
__global__ void k_e2(const float* __restrict__ emb, float* __restrict__ e2) {
    int j = threadIdx.x;   // blockDim = 64
    float s = 0.0f;
    for (int c = 0; c < NH; ++c) { float v = emb[j * NH + c]; s += v * v; }
    e2[j] = s;
}

// ---------------------------------------------------------------- finalize x (in place) + row stats
// one wave per row, 8 waves per block
__global__ __launch_bounds__(256) void k_fin(float* __restrict__ x, const int* __restrict__ cnt,
                                             float* __restrict__ row_sq, float* __restrict__ row_np) {
    int tid = threadIdx.x, w = tid >> 5, lane = tid & 31;
    int r = blockIdx.x * 8 + w;
    int c0 = cnt[r];
    float inv = 1.0f / (float)(c0 > 1 ? c0 : 1);
    float sq = 0.0f, ab = 0.0f;
    float* xr = x + (size_t)r * NH;
    for (int c = lane; c < NH; c += 32) {
        float v = xr[c] * inv;
        xr[c] = v;
        sq += v * v;
        ab += fabsf(v);
    }
    for (int off = 16; off; off >>= 1) {
        sq += __shfl_xor(sq, off, 32);
        ab += __shfl_xor(ab, off, 32);
    }
    if (lane == 0) {
        row_sq[r] = sq;
        row_np[r] = (ab > 0.0f) ? 1.0f : 0.0f;
    }
}

// ---------------------------------------------------------------- WMMA distances + argmin
// one wave = one 16-row tile; D(16x64) = X(16x256) @ E^T(256x64) via v_wmma_f32_16x16x4_f32
__global__ __launch_bounds__(64) void k_dist(const float* __restrict__ x, const float* __restrict__ emb,
                                             const float* __restrict__ e2, const float* __restrict__ row_sq,
                                             int* __restrict__ indices, int* __restrict__ hist) {
    __shared__ float dsbuf[2][16 * 68];          // padded stride to dodge bank conflicts
    int tid = threadIdx.x, w = tid >> 5, lane = tid & 31;
    int tile = blockIdx.x * 2 + w;
    int rowbase = tile * 16;
    int m = lane & 15;
    int koff = 2 * (lane >> 4);                  // K sub-offset per ISA A/B 16x4 f32 layout
    const float* xr = x + (size_t)(rowbase + m) * NH + koff;
    const float* e0 = emb + (size_t)(0 * 16 + m) * NH + koff;
    const float* e1 = emb + (size_t)(1 * 16 + m) * NH + koff;
    const float* e2p = emb + (size_t)(2 * 16 + m) * NH + koff;
    const float* e3 = emb + (size_t)(3 * 16 + m) * NH + koff;
    v8f c0 = {}, c1 = {}, c2 = {}, c3 = {};
    for (int kk = 0; kk < 64; ++kk) {
        v2f a  = *(const v2f*)(xr  + 4 * kk);
        v2f b0 = *(const v2f*)(e0  + 4 * kk);
        v2f b1 = *(const v2f*)(e1  + 4 * kk);
        v2f b2 = *(const v2f*)(e2p + 4 * kk);
        v2f b3 = *(const v2f*)(e3  + 4 * kk);
        c0 = __builtin_amdgcn_wmma_f32_16x16x4_f32(false, a, false, b0, (short)0, c0, false, false);
        c1 = __builtin_amdgcn_wmma_f32_16x16x4_f32(false, a, false, b1, (short)0, c1, false, false);
        c2 = __builtin_amdgcn_wmma_f32_16x16x4_f32(false, a, false, b2, (short)0, c2, false, false);
        c3 = __builtin_amdgcn_wmma_f32_16x16x4_f32(false, a, false, b3, (short)0, c3, false, false);
    }
    // C/D layout: VGPR i -> row (i + 8*(lane>=16)), col = lane%16 (+16*nt)
    int rhalf = 8 * (lane >> 4);
    for (int i = 0; i < 8; ++i) {
        dsbuf[w][(i + rhalf) * 68 +  0 + m] = c0[i];
        dsbuf[w][(i + rhalf) * 68 + 16 + m] = c1[i];
        dsbuf[w][(i + rhalf) * 68 + 32 + m] = c2[i];
        dsbuf[w][(i + rhalf) * 68 + 48 + m] = c3[i];
    }
    __syncthreads();
    if (lane < 16) {
        int r = rowbase + lane;
        float x2r = row_sq[r];
        float best = 3.4e38f;
        int bi = 0;
        for (int j = 0; j < NM; ++j) {
            float d = (e2[j] + x2r) - 2.0f * dsbuf[w][lane * 68 + j];
            if (d < best) { best = d; bi = j; }   // first-index tie-break, like jnp.argmin
        }
        indices[r] = bi;
        atomicAdd(&hist[bi], 1);
    }
}

// ---------------------------------------------------------------- z output + per-row latent loss
__global__ __launch_bounds__(256) void k_z(const float* __restrict__ x, const float* __restrict__ emb,
                                           const int* __restrict__ indices, const float* __restrict__ row_np,
                                           float* __restrict__ z, float* __restrict__ elat) {
    int tid = threadIdx.x, w = tid >> 5, lane = tid & 31;
    int r = blockIdx.x * 8 + w;
    int idx = indices[r];
    const float* xr = x + (size_t)r * NH;
    const float* qr = emb + (size_t)idx * NH;
    float* zr = z + (size_t)r * NH;
    float s = 0.0f;
    for (int c = lane; c < NH; c += 32) {
        float xv = xr[c], qv = qr[c];
        zr[c] = xv + (qv - xv);          // same association as reference
        float d = xv - qv;
        s += d * d;
    }
    for (int off = 16; off; off >>= 1) s += __shfl_xor(s, off, 32);
    if (lane == 0) elat[r] = (s / (float)NH) * row_np[r];
}

// ---------------------------------------------------------------- deterministic scalar reduce
__global__ void k_scalar(const int* __restrict__ hist, const float* __restrict__ elat,
                         const float* __restrict__ row_np, float* __restrict__ out) {
    float num = 0.0f, den = 0.0f;
    for (int r = 0; r < NROWS; ++r) { num += elat[r]; den += row_np[r]; }
    float ent = 0.0f;
    for (int j = 0; j < NM; ++j) {
        float p = (float)hist[j] / (float)NROWS;
        ent += p * logf(p + 1e-10f);
    }
    out[0] = 0.25f * num / den;   // COMMIT * sum(e_latent*np)/sum(np)
    out[1] = expf(-ent);          // perplexity
}

// ---------------------------------------------------------------- launch
extern "C" void kernel_launch(void* const* d_in, const int* in_sizes, int n_in,
                              void* d_out, int out_size, void* d_ws, size_t ws_size,
                              hipStream_t stream) {
    const float* mels   = (const float*)d_in[0];   // (16,8192,256) f32
    const int*   mel2ph = (const int*)d_in[1];     // (16,8192) i32
    const float* emb    = (const float*)d_in[2];   // (64,256) f32
    float* out = (float*)d_out;                    // z (16*512*256) + loss + perplexity

    char* ws = (char*)d_ws;
    size_t off = 0;
    float* X    = (float*)(ws + off); off += (size_t)NROWS * NH * sizeof(float);  // sums -> means (in place)
    int*   CNT  = (int*)  (ws + off); off += (size_t)NROWS * sizeof(int);
    float* RSQ  = (float*)(ws + off); off += (size_t)NROWS * sizeof(float);
    float* RNP  = (float*)(ws + off); off += (size_t)NROWS * sizeof(float);
    float* ELAT = (float*)(ws + off); off += (size_t)NROWS * sizeof(float);
    int*   IDX  = (int*)  (ws + off); off += (size_t)NROWS * sizeof(int);
    float* E2   = (float*)(ws + off); off += 256;
    int*   HIST = (int*)  (ws + off); off += 256;

    float* Z = out;
    float* SCAL = out + (size_t)NROWS * NH;

    k_init<<<1, 64, 0, stream>>>(HIST);
    k_cnt<<<NB, 256, 0, stream>>>(mel2ph, CNT);
    k_seg<<<NB * 8, 256, 0, stream>>>(mels, mel2ph, X);
    k_e2<<<1, 64, 0, stream>>>(emb, E2);
    k_fin<<<NROWS / 8, 256, 0, stream>>>(X, CNT, RSQ, RNP);
    k_dist<<<NTILES / 2, 64, 0, stream>>>(X, emb, E2, RSQ, IDX, HIST);
    k_z<<<NROWS / 8, 256, 0, stream>>>(X, emb, IDX, RNP, Z, ELAT);
    k_scalar<<<1, 1, 0, stream>>>(HIST, ELAT, RNP, SCAL);
}